// GNNEncoder_5566277616603
// MI455X (gfx1250) — compile-verified
//
#include <hip/hip_runtime.h>

#define D 128

typedef __attribute__((ext_vector_type(2))) float v2f;
typedef __attribute__((ext_vector_type(8))) float v8f;

__device__ __forceinline__ float lrelu(float v) { return v > 0.0f ? v : 0.01f * v; }

// ---------------- degree / norm ----------------
__global__ void k_fill1(float* __restrict__ p, int n) {
    int i = blockIdx.x * blockDim.x + threadIdx.x;
    if (i < n) p[i] = 1.0f;   // self-loop contributes 1 to every degree
}

__global__ void k_deg_edges(const int* __restrict__ colv, float* __restrict__ deg, int e) {
    int i = blockIdx.x * blockDim.x + threadIdx.x;
    if (i < e) atomicAdd(&deg[colv[i]], 1.0f);
}

__global__ void k_rsqrt(float* __restrict__ p, int n) {
    int i = blockIdx.x * blockDim.x + threadIdx.x;
    if (i < n) p[i] = rsqrtf(p[i]);   // deg >= 1 always (self-loops)
}

// ---------------- WMMA GEMM: H = act(A) @ W ----------------
// One workgroup (256 thr = 8 waves) computes a 16x128 tile of H.
// Wave w owns columns [16w, 16w+16). K-loop: 32 x v_wmma_f32_16x16x4_f32.
__global__ __launch_bounds__(256) void k_gemm_wmma(const float* __restrict__ A,
                                                   const float* __restrict__ W,
                                                   float* __restrict__ H,
                                                   int n, int act) {
    __shared__ float As[16 * D];   // 8 KB A-tile, row-major
    const int t    = threadIdx.x;
    const int row0 = blockIdx.x * 16;

    // Stage 16xD rows of A into LDS, coalesced float4, activation fused.
    {
        const float4* src = (const float4*)A;
        float4*       dst = (float4*)As;
        for (int i = t; i < 16 * (D / 4); i += 256) {       // 512 float4
            int r  = row0 + (i >> 5);                        // D/4 = 32 per row
            int rc = r < n ? r : (n - 1);                    // clamp (keep EXEC full)
            float4 v = src[(size_t)rc * (D / 4) + (i & 31)];
            if (act) { v.x = lrelu(v.x); v.y = lrelu(v.y); v.z = lrelu(v.z); v.w = lrelu(v.w); }
            dst[i] = v;
        }
    }
    __syncthreads();

    const int wave = t >> 5;          // 0..7 -> 16-column slice
    const int lane = t & 31;
    const int hi   = lane >> 4;       // selects K pair (0,1) vs (2,3)
    const int lm   = lane & 15;       // M index for A, N index for B/D
    const int col0 = wave * 16;

    const float* arow = As + lm * D;
    v8f acc = {};
    for (int k = 0; k < D; k += 4) {
        const int ks = k + hi * 2;
        // A 16x4 fp32 fragment: lanes 0-15 hold K=0,1; lanes 16-31 hold K=2,3
        v2f av = { arow[ks], arow[ks + 1] };
        // B 4x16 fp32 fragment: row-striped across lanes, K split by lane half
        v2f bv = { W[ks * D + col0 + lm], W[(ks + 1) * D + col0 + lm] };
        acc = __builtin_amdgcn_wmma_f32_16x16x4_f32(false, av, false, bv,
                                                    (short)0, acc, false, false);
    }
    // D 16x16 f32: VGPR j -> M = j + 8*hi, N = lm
#pragma unroll
    for (int j = 0; j < 8; ++j) {
        int r = row0 + hi * 8 + j;
        if (r < n) H[(size_t)r * D + col0 + lm] = acc[j];
    }
}

// ---------------- aggregation ----------------
// out[node] = bias + dis[node]^2 * h[node]   (self-loop term + bias folded in)
__global__ void k_init_agg(const float* __restrict__ h, const float* __restrict__ dis,
                           const float* __restrict__ bias, float* __restrict__ out, int n) {
    int i = blockIdx.x * blockDim.x + threadIdx.x;
    if (i < n * D) {
        int node = i >> 7;
        int d    = i & (D - 1);
        float s  = dis[node];
        out[i] = fmaf(s * s, h[i], bias[d]);
    }
}

// One wave per edge: 32 lanes x float4 covers the 128-wide feature row.
__global__ __launch_bounds__(256) void k_scatter(const float* __restrict__ h,
                                                 const float* __restrict__ dis,
                                                 const int* __restrict__ rowv,
                                                 const int* __restrict__ colv,
                                                 float* __restrict__ out, int e) {
    int eid = blockIdx.x * 8 + (threadIdx.x >> 5);
    if (eid >= e) return;
    int lane = threadIdx.x & 31;
    int r = rowv[eid], c = colv[eid];
    float nrm = dis[r] * dis[c];
    float4 v = ((const float4*)(h + (size_t)r * D))[lane];
    float* dst = out + (size_t)c * D + lane * 4;
    atomicAdd(dst + 0, v.x * nrm);
    atomicAdd(dst + 1, v.y * nrm);
    atomicAdd(dst + 2, v.z * nrm);
    atomicAdd(dst + 3, v.w * nrm);
}

// ---------------- pooling ----------------
__global__ void k_zero(float* __restrict__ out, int n, int* __restrict__ counts, int g) {
    int i = blockIdx.x * blockDim.x + threadIdx.x;
    if (i < n) out[i] = 0.0f;
    if (i < g) counts[i] = 0;
}

// One wave per node: segment-sum into graph embedding + per-graph node counts.
__global__ __launch_bounds__(256) void k_pool(const float* __restrict__ emb,
                                              const int* __restrict__ batch,
                                              float* __restrict__ gsum,
                                              int* __restrict__ counts, int n) {
    int node = blockIdx.x * 8 + (threadIdx.x >> 5);
    if (node >= n) return;
    int lane = threadIdx.x & 31;
    int g = batch[node];
    float4 v = ((const float4*)(emb + (size_t)node * D))[lane];
    float* dst = gsum + (size_t)g * D + lane * 4;
    atomicAdd(dst + 0, v.x);
    atomicAdd(dst + 1, v.y);
    atomicAdd(dst + 2, v.z);
    atomicAdd(dst + 3, v.w);
    if (lane == 0) atomicAdd(&counts[g], 1);
}

// Single block: exclusive scan of counts (= cum_num) then gather first-node rows.
__global__ void k_first(const float* __restrict__ emb, const int* __restrict__ counts,
                        float* __restrict__ out, int n, int g) {
    __shared__ int cum[256];
    if (threadIdx.x == 0) {
        int s = 0;
        for (int i = 0; i < g && i < 256; ++i) { cum[i] = s; s += counts[i]; }
    }
    __syncthreads();
    for (int i = 0; i < g && i < 256; ++i) {
        int f = cum[i];
        if (f >= n) f = n - 1;
        out[(size_t)i * D + threadIdx.x] = emb[(size_t)f * D + threadIdx.x];
    }
}

extern "C" void kernel_launch(void* const* d_in, const int* in_sizes, int n_in,
                              void* d_out, int out_size, void* d_ws, size_t ws_size,
                              hipStream_t stream) {
    (void)n_in; (void)ws_size;
    const float* x     = (const float*)d_in[0];
    const int*   edge  = (const int*)d_in[1];
    const int*   batch = (const int*)d_in[2];
    const float* Wm[3] = { (const float*)d_in[3], (const float*)d_in[5], (const float*)d_in[7] };
    const float* bm[3] = { (const float*)d_in[4], (const float*)d_in[6], (const float*)d_in[8] };

    const int N = in_sizes[2];          // batch has N entries
    const int E = in_sizes[1] / 2;      // edge_index is (2, E)
    const int G = out_size / (2 * D);   // tuple of two (G, D) outputs

    const int* rowv = edge;             // edge_index[0]
    const int* colv = edge + E;         // edge_index[1]

    // workspace layout
    float* hA     = (float*)d_ws;                  // N*D  (GEMM output)
    float* hB     = hA + (size_t)N * D;            // N*D  (aggregation output)
    float* dis    = hB + (size_t)N * D;            // N    (deg -> rsqrt(deg))
    int*   counts = (int*)(dis + N);               // G

    // degrees / symmetric norm
    k_fill1    <<<(N + 255) / 256, 256, 0, stream>>>(dis, N);
    k_deg_edges<<<(E + 255) / 256, 256, 0, stream>>>(colv, dis, E);
    k_rsqrt    <<<(N + 255) / 256, 256, 0, stream>>>(dis, N);

    const int gemmBlocks = (N + 15) / 16;
    const int edgeBlocks = (E + 7) / 8;
    const int nodeBlocks = (N + 7) / 8;
    const unsigned initBlocks = (unsigned)(((size_t)N * D + 255) / 256);

    const float* in = x;
    for (int l = 0; l < 3; ++l) {
        // GEMM with previous layer's leaky-relu fused into the A load
        k_gemm_wmma<<<gemmBlocks, 256, 0, stream>>>(in, Wm[l], hA, N, l > 0 ? 1 : 0);
        // bias + self-loop term
        k_init_agg<<<initBlocks, 256, 0, stream>>>(hA, dis, bm[l], hB, N);
        // edge scatter (L2-resident accumulator)
        k_scatter<<<edgeBlocks, 256, 0, stream>>>(hA, dis, rowv, colv, hB, E);
        in = hB;   // leaky-relu of hB applied inside next GEMM's staging
    }
    // hB now holds node_emb (layer 3 has no activation)

    float* firstOut = (float*)d_out;                  // (G, D) first-node rows
    float* gsum     = (float*)d_out + (size_t)G * D;  // (G, D) graph sums

    k_zero <<<(out_size + 255) / 256, 256, 0, stream>>>((float*)d_out, out_size, counts, G);
    k_pool <<<nodeBlocks, 256, 0, stream>>>(hB, batch, gsum, counts, N);
    k_first<<<1, D, 0, stream>>>(hB, counts, firstOut, N, G);
}